// TargetLinear_28647431864941
// MI455X (gfx1250) — compile-verified
//
#include <hip/hip_runtime.h>
#include <hip/hip_bf16.h>

// ---------------------------------------------------------------------------
// TargetLinear hypernetwork kernel for MI455X (gfx1250, wave32, WMMA).
//
// Sizes (fixed by setup_inputs): B=16 L=8192 -> M=131072 rows, Ts=64, H=8,
// Dh=256, R=32.  out[m,s] = sum_h mask[m,h] * (x[m,:] @ W[h])[s] + mask@bias.
//
// Restructured as one GEMM with K' = H*Ts + H = 520:
//   A'[m, h*64+d]  = mask[m,h] * x[m,d]      (built on the fly in VGPRs)
//   A'[m, 512+h]   = mask[m,h]               (bias rows)
//   B'[h*64+d, s]  = W[h,d,s] = sum_r down[h,d,r]*up[h,r,s]
//   B'[512+h, s]   = bias[h,s]
// computed with V_WMMA_F32_16X16X4_F32 (exact f32 accumulate, 130 K-steps).
// B' (133 KB) is staged once per block into LDS (320 KB/WGP on CDNA5),
// preferably via the async global->LDS copy path (ASYNCcnt).
// ---------------------------------------------------------------------------

typedef __attribute__((ext_vector_type(2))) float v2f;
typedef __attribute__((ext_vector_type(8))) float v8f;
typedef __attribute__((ext_vector_type(4))) int v4i;

#if defined(__has_builtin)
#if __has_builtin(__builtin_amdgcn_global_load_async_to_lds_b128) && \
    __has_builtin(__builtin_amdgcn_s_wait_asynccnt)
#define USE_ASYNC_LDS_COPY 1
#endif
#endif
#ifndef USE_ASYNC_LDS_COPY
#define USE_ASYNC_LDS_COPY 0
#endif

// B' is stored packed so that the per-lane WMMA B fragment (k,k+1 @ column n)
// is a single coalesced float2:   Wpk[(k>>1)*128 + n*2 + (k&1)] = B'[k][n]
#define WPK_IDX(k, n) (((k) >> 1) * 128 + (n) * 2 + ((k) & 1))

// ---------------------------------------------------------------------------
// Kernel 1: hypernetwork -> packed B' (520 x 64 f32 = 133,120 B) in d_ws.
// One block per h (8 blocks). Tiny: ~10 MFLOP total.
// ---------------------------------------------------------------------------
__global__ __launch_bounds__(256) void hyper_prep_kernel(
    const float* __restrict__ hyper,  // (8, 256)
    const float* __restrict__ W1,     // (256, 4096)
    const float* __restrict__ b1,     // (4096,)
    const float* __restrict__ Wb,     // (256, 64)
    const float* __restrict__ bb,     // (64,)
    float* __restrict__ Wpk)          // packed B' (520*64 floats)
{
    __shared__ float hyp[256];
    __shared__ float hw[4096];
    const int tid = threadIdx.x;
    const int h = blockIdx.x;

    hyp[tid] = hyper[h * 256 + tid];
    __syncthreads();

    // hw = hyper[h] @ W1 + b1   (each thread: 16 of 4096 outputs, coalesced j)
    for (int i = 0; i < 16; ++i) {
        const int j = tid + 256 * i;
        float s = b1[j];
        for (int d = 0; d < 256; ++d)
            s = fmaf(hyp[d], W1[d * 4096 + j], s);
        hw[j] = s;
    }
    __syncthreads();

    // W[h][t][s] = sum_r down[t][r]*up[r][s]; down=hw[:2048], up=hw[2048:]
    for (int i = 0; i < 16; ++i) {
        const int idx = tid + 256 * i;
        const int t = idx >> 6;
        const int s = idx & 63;
        float acc = 0.0f;
#pragma unroll
        for (int r = 0; r < 32; ++r)
            acc = fmaf(hw[t * 32 + r], hw[2048 + r * 64 + s], acc);
        const int k = h * 64 + t;
        Wpk[WPK_IDX(k, s)] = acc;
    }

    // bias row: B'[512+h][s] = hyper[h] @ Wb + bb
    if (tid < 64) {
        const int s = tid;
        float acc = bb[s];
        for (int d = 0; d < 256; ++d)
            acc = fmaf(hyp[d], Wb[d * 64 + s], acc);
        const int k = 512 + h;
        Wpk[WPK_IDX(k, s)] = acc;
    }
}

// ---------------------------------------------------------------------------
// Kernel 2: main masked batched-linear as one WMMA f32 GEMM.
// 256 threads = 8 wave32; each wave owns 32 rows (two 16-row WMMA tiles).
// grid = 131072 / 256 = 512 blocks. No divergence at WMMA -> EXEC all-ones.
// B' lives in LDS (16640 float2 = 133,120 B; CDNA5 WGP has 320 KB).
// ---------------------------------------------------------------------------
__global__ __launch_bounds__(256) void target_linear_wmma_kernel(
    const float* __restrict__ X,     // (131072, 64)
    const float* __restrict__ mask,  // (131072, 8)
    const float* __restrict__ Wpk,   // packed B' (520 x 64)
    float* __restrict__ out)         // (131072, 64)
{
    __shared__ float2 Wlds[16640];  // packed B', float2 per (kpair, n)

    // ---- stage B' into LDS (8320 x b128), async path when available ----
    {
        const float4* __restrict__ gsrc = (const float4*)Wpk;
        float4* ldst = (float4*)Wlds;
#if USE_ASYNC_LDS_COPY
        for (int i = threadIdx.x; i < 8320; i += 256) {
            __builtin_amdgcn_global_load_async_to_lds_b128(
                (v4i*)(gsrc + i), (v4i*)(ldst + i), 0, 0);
        }
        __builtin_amdgcn_s_wait_asynccnt(0);
#else
        for (int i = threadIdx.x; i < 8320; i += 256)
            ldst[i] = gsrc[i];
#endif
        __syncthreads();
    }

    const int lane = threadIdx.x & 31;
    const int wave = threadIdx.x >> 5;
    const int m = lane & 15;   // row-within-tile for A; column for B/C
    const int kh = lane >> 4;  // K-half select (A: K {0,1} vs {2,3})
    const int rowbase = (blockIdx.x * 8 + wave) * 32;

    // Preload this wave's X rows once; reused across all 8 h-slices.
    // Lane needs x[row, d] for d = 4*kk + 2*kh + {0,1}  -> 16 float2 per tile.
    float2 xa0[16], xa1[16];
    {
        const float* xp0 = X + (rowbase + m) * 64 + 2 * kh;
        const float* xp1 = X + (rowbase + 16 + m) * 64 + 2 * kh;
#pragma unroll
        for (int kk = 0; kk < 16; ++kk) {
            xa0[kk] = *(const float2*)(xp0 + kk * 4);
            xa1[kk] = *(const float2*)(xp1 + kk * 4);
        }
    }

    v8f acc[2][4] = {};  // 2 row tiles x 4 N-tiles, f32 16x16 accumulators

    // ---- main K region: k = h*64 + d, A' = mask[m,h]*x[m,d] ----
#pragma unroll 1
    for (int h = 0; h < 8; ++h) {
        const float mv0 = mask[(rowbase + m) * 8 + h];
        const float mv1 = mask[(rowbase + 16 + m) * 8 + h];
#pragma unroll
        for (int kk = 0; kk < 16; ++kk) {
            v2f a0, a1;
            a0.x = mv0 * xa0[kk].x; a0.y = mv0 * xa0[kk].y;
            a1.x = mv1 * xa1[kk].x; a1.y = mv1 * xa1[kk].y;
            const int kpair = h * 32 + kk * 2;  // (h*64 + kk*4) >> 1
#pragma unroll
            for (int nt = 0; nt < 4; ++nt) {
                const float2 bf = Wlds[(kpair + kh) * 64 + nt * 16 + m];
                v2f b; b.x = bf.x; b.y = bf.y;
                acc[0][nt] = __builtin_amdgcn_wmma_f32_16x16x4_f32(
                    false, a0, false, b, (short)0, acc[0][nt], false, false);
                acc[1][nt] = __builtin_amdgcn_wmma_f32_16x16x4_f32(
                    false, a1, false, b, (short)0, acc[1][nt], false, false);
            }
        }
    }

    // ---- bias K region: k = 512..519, A' = mask[m, k-512], B' = bias ----
#pragma unroll
    for (int kp = 0; kp < 2; ++kp) {
        const float2 mv20 =
            *(const float2*)(mask + (rowbase + m) * 8 + kp * 4 + 2 * kh);
        const float2 mv21 =
            *(const float2*)(mask + (rowbase + 16 + m) * 8 + kp * 4 + 2 * kh);
        v2f a0, a1;
        a0.x = mv20.x; a0.y = mv20.y;
        a1.x = mv21.x; a1.y = mv21.y;
        const int kpair = 256 + kp * 2;  // (512 + kp*4) >> 1
#pragma unroll
        for (int nt = 0; nt < 4; ++nt) {
            const float2 bf = Wlds[(kpair + kh) * 64 + nt * 16 + m];
            v2f b; b.x = bf.x; b.y = bf.y;
            acc[0][nt] = __builtin_amdgcn_wmma_f32_16x16x4_f32(
                false, a0, false, b, (short)0, acc[0][nt], false, false);
            acc[1][nt] = __builtin_amdgcn_wmma_f32_16x16x4_f32(
                false, a1, false, b, (short)0, acc[1][nt], false, false);
        }
    }

    // ---- store: C layout VGPR r -> row r (lanes 0-15) / r+8 (lanes 16-31)
#pragma unroll
    for (int rt = 0; rt < 2; ++rt) {
#pragma unroll
        for (int nt = 0; nt < 4; ++nt) {
#pragma unroll
            for (int r = 0; r < 8; ++r) {
                out[(rowbase + rt * 16 + r + 8 * kh) * 64 + nt * 16 + m] =
                    acc[rt][nt][r];
            }
        }
    }
}

// ---------------------------------------------------------------------------
extern "C" void kernel_launch(void* const* d_in, const int* in_sizes, int n_in,
                              void* d_out, int out_size, void* d_ws, size_t ws_size,
                              hipStream_t stream) {
    const float* target_ipt = (const float*)d_in[0];  // (16, 8192, 64)
    const float* hyper_ipt  = (const float*)d_in[1];  // (8, 256)
    const float* hyp_mask   = (const float*)d_in[2];  // (16, 8192, 8, 1)
    const float* W1         = (const float*)d_in[3];  // (256, 4096)
    const float* b1         = (const float*)d_in[4];  // (4096,)
    const float* Wb         = (const float*)d_in[5];  // (256, 64)
    const float* bb         = (const float*)d_in[6];  // (64,)
    float* out = (float*)d_out;
    float* Wpk = (float*)d_ws;  // 520*64*4 = 133,120 bytes of scratch

    (void)in_sizes; (void)n_in; (void)out_size; (void)ws_size;

    hyper_prep_kernel<<<8, 256, 0, stream>>>(hyper_ipt, W1, b1, Wb, bb, Wpk);

    // M = 131072 rows, 256 rows per block -> 512 blocks
    target_linear_wmma_kernel<<<512, 256, 0, stream>>>(target_ipt, hyp_mask,
                                                       Wpk, out);
}